// EgoGNN_87479893885153
// MI455X (gfx1250) — compile-verified
//
#include <hip/hip_runtime.h>
#include <hip/hip_bf16.h>
#include <math.h>

typedef float v2f __attribute__((ext_vector_type(2)));
typedef float v8f __attribute__((ext_vector_type(8)));

#define FEAT 128

// ---------------------------------------------------------------- elementwise
__global__ void zero_kernel(float* __restrict__ p, int n4) {
    int i = blockIdx.x * blockDim.x + threadIdx.x;
    if (i < n4) *(float4*)(p + (long long)i * 4) = make_float4(0.f, 0.f, 0.f, 0.f);
}

__global__ void copy_kernel(const float* __restrict__ src, float* __restrict__ dst, int n4) {
    int i = blockIdx.x * blockDim.x + threadIdx.x;
    if (i < n4) *(float4*)(dst + (long long)i * 4) = *(const float4*)(src + (long long)i * 4);
}

// Wt[n*K + k] = W[k*N + n]   (weights are tiny: <= 64 KB)
__global__ void transpose_kernel(const float* __restrict__ W, float* __restrict__ Wt,
                                 int K, int N) {
    int idx = blockIdx.x * blockDim.x + threadIdx.x;
    if (idx < K * N) {
        int k = idx / N, n = idx - k * N;
        Wt[n * K + k] = W[idx];
    }
}

// ---------------------------------------------------------------- scatter-add
// out[dst[e]][:] += x[src[e]][:]  (one wave per edge; indices fetched scalar)
__global__ __launch_bounds__(256) void scatter_add_kernel(
    const float* __restrict__ x, const int* __restrict__ dst,
    const int* __restrict__ src, float* __restrict__ out, int nE) {
    int e = __builtin_amdgcn_readfirstlane(blockIdx.x * 8 + (threadIdx.x >> 5));
    if (e >= nE) return;
    int lane = threadIdx.x & 31;
    int s = src[e];   // wave-uniform -> scalar load
    int d = dst[e];
    float4 v = *(const float4*)(x + (long long)s * FEAT + lane * 4);
    float* o = out + (long long)d * FEAT + lane * 4;
    unsafeAtomicAdd(o + 0, v.x);
    unsafeAtomicAdd(o + 1, v.y);
    unsafeAtomicAdd(o + 2, v.z);
    unsafeAtomicAdd(o + 3, v.w);
}

// ---------------------------------------------------------------- GEMM 128x128
// Y = act(scale * (A @ W) + bias), A:[M,128], Wt = W^T : [128(n),128(k)].
// Block: 256 thr = 8 waves; block owns 32 rows (LDS), wave w owns cols [16w,16w+16)
// with TWO 16x16 accumulators sharing each B fragment (halves B traffic).
__global__ __launch_bounds__(256) void gemm128_relu_kernel(
    const float* __restrict__ A, const float* __restrict__ Wt,
    const float* __restrict__ bias, float* __restrict__ Y,
    int M, float scale, int do_relu) {
    __shared__ float As[32 * FEAT];   // 16 KB

    const long long mb = (long long)blockIdx.x * 32;
    const int rowsLeft = (int)(M - mb);          // >= 16 always (M % 16 == 0)

    for (int i = threadIdx.x; i < (32 * FEAT) / 4; i += blockDim.x) {
        int rr = i >> 5;                         // row in tile (4 floats per i)
        int kk = (i * 4) & (FEAT - 1);
        long long gr = mb + (rr < rowsLeft ? rr : 0);   // clamp for tail block
        *(float4*)&As[rr * FEAT + kk] = *(const float4*)&A[gr * FEAT + kk];
    }
    __syncthreads();

    const int wave = threadIdx.x >> 5;
    const int lane = threadIdx.x & 31;
    const int nb   = wave * 16;
    const int r    = lane & 15;     // M within tile (A) / N within tile (B, C)
    const int hi   = lane >> 4;     // K-pair select {0,1} vs {2,3}
    const int col  = nb + r;

    v8f acc0 = {}, acc1 = {};
    const float* a0row = &As[r * FEAT];
    const float* a1row = &As[(r + 16) * FEAT];
    const float* btrow = &Wt[col * FEAT];
#pragma unroll
    for (int k0 = 0; k0 < FEAT; k0 += 4) {
        const int kb = k0 + 2 * hi;
        v2f a0 = *(const v2f*)&a0row[kb];
        v2f a1 = *(const v2f*)&a1row[kb];
        v2f b  = *(const v2f*)&btrow[kb];        // single b64 load, shared
        acc0 = __builtin_amdgcn_wmma_f32_16x16x4_f32(
            false, a0, false, b, (short)0, acc0, false, false);
        acc1 = __builtin_amdgcn_wmma_f32_16x16x4_f32(
            false, a1, false, b, (short)0, acc1, false, false);
    }

    const float bv = bias[col];
#pragma unroll
    for (int i = 0; i < 8; ++i) {
        long long row = mb + i + 8 * hi;         // C layout: M = vgpr + 8*(lane>=16)
        float v = acc0[i] * scale + bv;
        if (do_relu) v = v > 0.f ? v : 0.f;
        Y[row * FEAT + col] = v;                 // first tile always valid
    }
#pragma unroll
    for (int i = 0; i < 8; ++i) {
        long long row = mb + 16 + i + 8 * hi;
        if (row < M) {
            float v = acc1[i] * scale + bv;
            if (do_relu) v = v > 0.f ? v : 0.f;
            Y[row * FEAT + col] = v;
        }
    }
}

// ---------------------------------------------------------------- GEMM 128x40
// Y = A @ W + bias, Wt = W^T : [40(n),128(k)]. 3 waves/block, N padded 40->48.
__global__ __launch_bounds__(96) void gemm40_kernel(
    const float* __restrict__ A, const float* __restrict__ Wt,
    const float* __restrict__ bias, float* __restrict__ Y) {
    __shared__ float As[16 * FEAT];

    const long long mb = (long long)blockIdx.x * 16;
    for (int i = threadIdx.x; i < (16 * FEAT) / 4; i += blockDim.x)
        *(float4*)&As[i * 4] = *(const float4*)&A[mb * FEAT + i * 4];
    __syncthreads();

    const int wave = threadIdx.x >> 5;
    const int lane = threadIdx.x & 31;
    const int nb   = wave * 16;
    const int r    = lane & 15;
    const int hi   = lane >> 4;
    const int col  = nb + r;
    const bool valid = col < 40;
    const float* btrow = &Wt[(valid ? col : 0) * FEAT];

    v8f acc = {};
    const float* arow = &As[r * FEAT];
#pragma unroll
    for (int k0 = 0; k0 < FEAT; k0 += 4) {
        const int kb = k0 + 2 * hi;
        v2f a = *(const v2f*)&arow[kb];
        v2f b = *(const v2f*)&btrow[kb];
        if (!valid) { b.x = 0.f; b.y = 0.f; }
        acc = __builtin_amdgcn_wmma_f32_16x16x4_f32(
            false, a, false, b, (short)0, acc, false, false);
    }

    if (valid) {
        const float bv = bias[col];
#pragma unroll
        for (int i = 0; i < 8; ++i) {
            long long row = mb + i + 8 * hi;
            Y[row * 40 + col] = acc[i] + bv;
        }
    }
}

// ---------------------------------------------------------------- log_softmax (40-wide, in place)
__global__ void logsoftmax40_kernel(float* __restrict__ y, int M) {
    int row = blockIdx.x * blockDim.x + threadIdx.x;
    if (row >= M) return;
    float* p = y + (long long)row * 40;
    float m = -INFINITY;
#pragma unroll
    for (int i = 0; i < 40; ++i) m = fmaxf(m, p[i]);
    float s = 0.f;
#pragma unroll
    for (int i = 0; i < 40; ++i) s += __expf(p[i] - m);
    float ls = __logf(s);
#pragma unroll
    for (int i = 0; i < 40; ++i) p[i] = p[i] - m - ls;
}

// ---------------------------------------------------------------- driver
extern "C" void kernel_launch(void* const* d_in, const int* in_sizes, int n_in,
                              void* d_out, int out_size, void* d_ws, size_t ws_size,
                              hipStream_t stream) {
    const float* x_in = (const float*)d_in[0];
    const int*   ei   = (const int*)d_in[1];   // [2, E_REG]: row0 = src, row1 = dst (GIN)
    const int*   ego  = (const int*)d_in[2];   // [2, E_EGO]: row0 = dst, row1 = src
    const float* W1   = (const float*)d_in[3];
    const float* b1   = (const float*)d_in[4];
    const float* Wg1  = (const float*)d_in[5];
    const float* bg1  = (const float*)d_in[6];
    const float* W2   = (const float*)d_in[7];
    const float* b2   = (const float*)d_in[8];
    const float* Wg2  = (const float*)d_in[9];
    const float* bg2  = (const float*)d_in[10];
    float* out = (float*)d_out;

    const int n    = in_sizes[0] / FEAT;   // 50000
    const int Ereg = in_sizes[1] / 2;      // 800000
    const int Eego = in_sizes[2] / 2;      // 1600000
    const float inv_n = 1.0f / (float)n;

    float* buf0  = (float*)d_ws;
    float* buf1  = buf0 + (size_t)n * FEAT;
    float* Wt1   = buf1 + (size_t)n * FEAT;
    float* Wtg1  = Wt1 + FEAT * FEAT;
    float* Wt2   = Wtg1 + FEAT * FEAT;
    float* Wtg2  = Wt2 + FEAT * FEAT;      // 40 * 128

    const int NF4   = (n * FEAT) / 4;
    const int zgrid = (NF4 + 255) / 256;
    const int sgE   = (Eego + 7) / 8;      // 8 edges (waves) per 256-thr block
    const int sgR   = (Ereg + 7) / 8;
    const int mt32  = (n + 31) / 32;       // 32-row GEMM tiles
    const int mt16  = n / 16;              // exact for final GEMM

    const int* egoDst = ego;               // ego_ei[0] = segment (destination)
    const int* egoSrc = ego + Eego;        // ego_ei[1] = gather column
    const int* ginSrc = ei;                // x[ei[0]]
    const int* ginDst = ei + Ereg;         // segment ei[1]

    // One-time (per launch) weight transposes so B fragments load as b64.
    transpose_kernel<<<(FEAT * FEAT + 255) / 256, 256, 0, stream>>>(W1, Wt1, FEAT, FEAT);
    transpose_kernel<<<(FEAT * FEAT + 255) / 256, 256, 0, stream>>>(Wg1, Wtg1, FEAT, FEAT);
    transpose_kernel<<<(FEAT * FEAT + 255) / 256, 256, 0, stream>>>(W2, Wt2, FEAT, FEAT);
    transpose_kernel<<<(FEAT * 40 + 255) / 256, 256, 0, stream>>>(Wg2, Wtg2, FEAT, 40);

    // Layer 1: ego conv -> Linear(relu, fused 1/n) -> GIN -> Linear(relu)
    zero_kernel<<<zgrid, 256, 0, stream>>>(buf0, NF4);
    scatter_add_kernel<<<sgE, 256, 0, stream>>>(x_in, egoDst, egoSrc, buf0, Eego);
    gemm128_relu_kernel<<<mt32, 256, 0, stream>>>(buf0, Wt1, b1, buf1, n, inv_n, 1);

    copy_kernel<<<zgrid, 256, 0, stream>>>(buf1, buf0, NF4);
    scatter_add_kernel<<<sgR, 256, 0, stream>>>(buf1, ginDst, ginSrc, buf0, Ereg);
    gemm128_relu_kernel<<<mt32, 256, 0, stream>>>(buf0, Wtg1, bg1, buf1, n, 1.0f, 1);

    // Layer 2: ego conv -> Linear(relu, fused 1/n) -> GIN -> Linear(40) -> log_softmax
    zero_kernel<<<zgrid, 256, 0, stream>>>(buf0, NF4);
    scatter_add_kernel<<<sgE, 256, 0, stream>>>(buf1, egoDst, egoSrc, buf0, Eego);
    gemm128_relu_kernel<<<mt32, 256, 0, stream>>>(buf0, Wt2, b2, buf1, n, inv_n, 1);

    copy_kernel<<<zgrid, 256, 0, stream>>>(buf1, buf0, NF4);
    scatter_add_kernel<<<sgR, 256, 0, stream>>>(buf1, ginDst, ginSrc, buf0, Ereg);
    gemm40_kernel<<<mt16, 96, 0, stream>>>(buf0, Wtg2, bg2, out);

    logsoftmax40_kernel<<<(n + 255) / 256, 256, 0, stream>>>(out, n);
}